// LinearQuant_64450279244128
// MI455X (gfx1250) — compile-verified
//
#include <hip/hip_runtime.h>

// ---------------------------------------------------------------------------
// Fixed-point saturating quantization, Q4.4 (BW=8 bits, FL=4 fractional):
//   q = clip(floor(x * 16 + 0.5), -128, 127) * 0.0625
// Pure streaming elementwise op: 411 MB HBM traffic, ~0.26 GFLOP.
// Memory-bound by ~3 orders of magnitude -> optimize the VMEM path only:
//   * b128 loads/stores, 4 in flight per wave (64 B / thread / block-tile)
//   * non-temporal cache hints (CDNA5 TH=NT) since data is touched once and
//     the 411 MB working set exceeds the 192 MB L2
// ---------------------------------------------------------------------------

typedef float v4f __attribute__((ext_vector_type(4)));

__device__ __forceinline__ float quant1(float x) {
    // round-half-up via floor(x/delta + 0.5), then saturate to [-2^7, 2^7-1]
    float q = __builtin_floorf(__builtin_fmaf(x, 16.0f, 0.5f));
    q = __builtin_fmaxf(q, -128.0f);
    q = __builtin_fminf(q,  127.0f);
    return q * 0.0625f;
}

__device__ __forceinline__ v4f quant4(v4f x) {
    v4f r;
    r.x = quant1(x.x);
    r.y = quant1(x.y);
    r.z = quant1(x.z);
    r.w = quant1(x.w);
    return r;
}

// Each block owns 1024 consecutive float4s (16 KB in / 16 KB out).
// Thread t handles float4 indices blockBase + t + k*256, k = 0..3:
// every global_load_b128 in the wave is a fully-coalesced 512 B burst,
// and each wave keeps 4 b128 loads outstanding (LOADcnt depth 4).
__global__ __launch_bounds__(256) void
quant_vec4_kernel(const v4f* __restrict__ in, v4f* __restrict__ out, int n4) {
    const int blockBase = blockIdx.x * (256 * 4);
    const int t = threadIdx.x;

    if (blockBase + 256 * 4 <= n4) {
        // Fast path: whole tile in range, no branches between the 4 loads so
        // the compiler clusters them before the first s_wait_loadcnt.
        v4f x0 = __builtin_nontemporal_load(&in[blockBase + t          ]);
        v4f x1 = __builtin_nontemporal_load(&in[blockBase + t + 256    ]);
        v4f x2 = __builtin_nontemporal_load(&in[blockBase + t + 256 * 2]);
        v4f x3 = __builtin_nontemporal_load(&in[blockBase + t + 256 * 3]);
        __builtin_nontemporal_store(quant4(x0), &out[blockBase + t          ]);
        __builtin_nontemporal_store(quant4(x1), &out[blockBase + t + 256    ]);
        __builtin_nontemporal_store(quant4(x2), &out[blockBase + t + 256 * 2]);
        __builtin_nontemporal_store(quant4(x3), &out[blockBase + t + 256 * 3]);
    } else {
        #pragma unroll
        for (int k = 0; k < 4; ++k) {
            const int i = blockBase + t + k * 256;
            if (i < n4) {
                v4f x = __builtin_nontemporal_load(&in[i]);
                __builtin_nontemporal_store(quant4(x), &out[i]);
            }
        }
    }
}

// Scalar tail for n % 4 leftover elements (not taken for the reference shape,
// which is divisible by 16, but keeps the launcher shape-generic).
__global__ __launch_bounds__(32) void
quant_tail_kernel(const float* __restrict__ in, float* __restrict__ out,
                  int start, int n) {
    const int i = start + (int)threadIdx.x;
    if (i < n) {
        out[i] = quant1(in[i]);
    }
}

extern "C" void kernel_launch(void* const* d_in, const int* in_sizes, int n_in,
                              void* d_out, int out_size, void* d_ws, size_t ws_size,
                              hipStream_t stream) {
    (void)n_in; (void)out_size; (void)d_ws; (void)ws_size;

    const float* in  = (const float*)d_in[0];
    float*       out = (float*)d_out;
    const int n  = in_sizes[0];          // 51,380,224 for the reference shape
    const int n4 = n >> 2;               // float4 count
    const int rem = n & 3;

    if (n4 > 0) {
        const int vecPerBlock = 256 * 4; // 1024 float4s per block
        const int blocks = (n4 + vecPerBlock - 1) / vecPerBlock; // 12,544
        quant_vec4_kernel<<<blocks, 256, 0, stream>>>(
            (const v4f*)in, (v4f*)out, n4);
    }
    if (rem > 0) {
        quant_tail_kernel<<<1, 32, 0, stream>>>(in, out, n - rem, n);
    }
}